// Model_7017976562000
// MI455X (gfx1250) — compile-verified
//
#include <hip/hip_runtime.h>

#define N_NODES 50000
#define N_EDGES 800000
#define IN_CH   128
#define HID_CH  128
#define OUT_CH  64

typedef float v2f __attribute__((ext_vector_type(2)));
typedef float v8f __attribute__((ext_vector_type(8)));

// ---------------------------------------------------------------------------
// Dense GEMM using V_WMMA_F32_16X16X4_F32 (wave32, fp32), with fused epilogue:
//   H[row,col]   = (X @ W)[row,col]                       (needed by scatter)
//   AGG[row,col] = H[row,col] * dis[row]^2 + bias[col]    (self-loop + bias)
// One wave computes one 16x16 output tile; blockDim = (32, FOUT/16).
// A layout (16x4 f32): lane<16 holds K={k0,k0+1}, lane>=16 holds K={k0+2,k0+3}.
// B layout (4x16 f32): symmetric over N. C/D: 8 VGPRs, M=v (lanes>=16: M=v+8).
// ---------------------------------------------------------------------------
template<int K, int FOUT>
__global__ __launch_bounds__(32 * (FOUT / 16))
void gemm_wmma(const float* __restrict__ X, const float* __restrict__ W,
               const float* __restrict__ dis, const float* __restrict__ bias,
               float* __restrict__ H, float* __restrict__ AGG) {
  const int lane = threadIdx.x;          // 0..31
  const int col0 = threadIdx.y * 16;     // column tile of this wave
  const int row0 = blockIdx.x * 16;      // row tile of this block
  const int half = lane >> 4;            // 0: lanes 0-15, 1: lanes 16-31
  const int l    = lane & 15;

  const float* xp = X + (size_t)(row0 + l) * K + 2 * half;        // A pair
  const float* wp = W + (size_t)(2 * half) * FOUT + col0 + l;     // B pair

  v8f acc = {};
#pragma unroll
  for (int k = 0; k < K; k += 4) {
    v2f a = *(const v2f*)xp;     // X[row0+l][k+2h], X[row0+l][k+2h+1]
    v2f b;
    b.x = wp[0];                 // W[k+2h  ][col0+l]
    b.y = wp[FOUT];              // W[k+2h+1][col0+l]
    acc = __builtin_amdgcn_wmma_f32_16x16x4_f32(
        /*neg_a=*/false, a, /*neg_b=*/false, b,
        /*c_mod=*/(short)0, acc, /*reuse_a=*/false, /*reuse_b=*/false);
    xp += 4;
    wp += 4 * FOUT;
  }

  const float bcol = bias[col0 + l];
  const int rbase = row0 + half * 8;
  float* hp = H   + (size_t)rbase * FOUT + col0 + l;
  float* ap = AGG + (size_t)rbase * FOUT + col0 + l;
#pragma unroll
  for (int v = 0; v < 8; ++v) {
    float hval = acc[v];
    float di   = dis[rbase + v];              // row-uniform, L1 broadcast
    hp[(size_t)v * FOUT] = hval;
    ap[(size_t)v * FOUT] = hval * (di * di) + bcol;
  }
}

// ---------------------------------------------------------------------------
__global__ void fill_f32(float* __restrict__ p, float v, size_t n) {
  size_t i = (size_t)blockIdx.x * blockDim.x + threadIdx.x;
  if (i < n) p[i] = v;
}

// deg[dst] += 1 for every edge (deg pre-filled with 1.0 for self-loop)
__global__ void degree_count(const long long* __restrict__ dstIdx,
                             float* __restrict__ deg) {
  int e = blockIdx.x * blockDim.x + threadIdx.x;
  if (e < N_EDGES) unsafeAtomicAdd(&deg[(int)dstIdx[e]], 1.0f);
}

__global__ void compute_dis(const float* __restrict__ deg,
                            float* __restrict__ dis) {
  int n = blockIdx.x * blockDim.x + threadIdx.x;
  if (n < N_NODES) dis[n] = rsqrtf(deg[n]);
}

// ---------------------------------------------------------------------------
// One wave per edge: agg[dst] += h[src] * (dis[src]*dis[dst])
// F=128 -> float4 per lane; F=64 -> float2 per lane (all lanes busy).
// ---------------------------------------------------------------------------
template<int F>
__global__ void scatter_edges(const long long* __restrict__ srcIdx,
                              const long long* __restrict__ dstIdx,
                              const float* __restrict__ dis,
                              const float* __restrict__ h,
                              float* __restrict__ agg) {
  int wave = blockIdx.x * (blockDim.x >> 5) + (threadIdx.x >> 5);
  int lane = threadIdx.x & 31;
  if (wave >= N_EDGES) return;

  int s = (int)srcIdx[wave];
  int d = (int)dstIdx[wave];
  float norm = dis[s] * dis[d];
  const float* hs = h + (size_t)s * F;
  float* ad = agg + (size_t)d * F;

  constexpr int EPL = F / 32;  // elements per lane
  int f = lane * EPL;
  if constexpr (EPL == 4) {
    float4 v = *(const float4*)(hs + f);
    unsafeAtomicAdd(ad + f + 0, v.x * norm);
    unsafeAtomicAdd(ad + f + 1, v.y * norm);
    unsafeAtomicAdd(ad + f + 2, v.z * norm);
    unsafeAtomicAdd(ad + f + 3, v.w * norm);
  } else {
    float2 v = *(const float2*)(hs + f);
    unsafeAtomicAdd(ad + f + 0, v.x * norm);
    unsafeAtomicAdd(ad + f + 1, v.y * norm);
  }
}

// dst = relu(src), elementwise (in-place safe)
__global__ void relu_kernel(const float* __restrict__ src,
                            float* __restrict__ dst, size_t n) {
  size_t i = (size_t)blockIdx.x * blockDim.x + threadIdx.x;
  if (i < n) {
    float v = src[i];
    dst[i] = v > 0.0f ? v : 0.0f;
  }
}

// ---------------------------------------------------------------------------
extern "C" void kernel_launch(void* const* d_in, const int* in_sizes, int n_in,
                              void* d_out, int out_size, void* d_ws, size_t ws_size,
                              hipStream_t stream) {
  (void)in_sizes; (void)n_in; (void)out_size; (void)ws_size;

  const float*     x   = (const float*)d_in[0];
  const long long* ei  = (const long long*)d_in[1];   // int64 [2][E]
  const float*     W1  = (const float*)d_in[2];
  const float*     b1  = (const float*)d_in[3];
  const float*     W2  = (const float*)d_in[4];
  const float*     b2  = (const float*)d_in[5];
  float*           out = (float*)d_out;               // [N, 64]

  // workspace layout (floats): deg | dis | h1 | agg1 | h2 | agg2  (~77 MB)
  float* deg  = (float*)d_ws;
  float* dis  = deg  + N_NODES;
  float* h1   = dis  + N_NODES;
  float* agg1 = h1   + (size_t)N_NODES * HID_CH;
  float* h2   = agg1 + (size_t)N_NODES * HID_CH;
  float* agg2 = h2   + (size_t)N_NODES * OUT_CH;

  const long long* srcIdx = ei;
  const long long* dstIdx = ei + N_EDGES;

  const size_t n1 = (size_t)N_NODES * HID_CH;
  const size_t n2 = (size_t)N_NODES * OUT_CH;

  // shared graph normalization: deg = 1 + in-degree; dis = deg^-1/2
  fill_f32<<<(N_NODES + 255) / 256, 256, 0, stream>>>(deg, 1.0f, N_NODES);
  degree_count<<<(N_EDGES + 255) / 256, 256, 0, stream>>>(dstIdx, deg);
  compute_dis<<<(N_NODES + 255) / 256, 256, 0, stream>>>(deg, dis);

  // ---- layer 1: 128 -> 128 ----
  // gemm writes h1 and initializes agg1 = h1/deg + b1 (self-loop + bias)
  gemm_wmma<IN_CH, HID_CH>
      <<<N_NODES / 16, dim3(32, HID_CH / 16), 0, stream>>>(x, W1, dis, b1, h1, agg1);
  scatter_edges<HID_CH>
      <<<(N_EDGES + 7) / 8, 256, 0, stream>>>(srcIdx, dstIdx, dis, h1, agg1);
  relu_kernel<<<(int)((n1 + 255) / 256), 256, 0, stream>>>(agg1, agg1, n1);

  // ---- layer 2: 128 -> 64 ----
  gemm_wmma<HID_CH, OUT_CH>
      <<<N_NODES / 16, dim3(32, OUT_CH / 16), 0, stream>>>(agg1, W2, dis, b2, h2, agg2);
  scatter_edges<OUT_CH>
      <<<(N_EDGES + 7) / 8, 256, 0, stream>>>(srcIdx, dstIdx, dis, h2, agg2);
  relu_kernel<<<(int)((n2 + 255) / 256), 256, 0, stream>>>(agg2, out, n2);
}